// DMLHead_58033598103655
// MI455X (gfx1250) — compile-verified
//
#include <hip/hip_runtime.h>

typedef __attribute__((ext_vector_type(2))) float v2f;
typedef __attribute__((ext_vector_type(8))) float v8f;

#define HW    16384      // 128*128
#define EDIM  256
#define NCLS  81
#define NPAD  96         // 6 WMMA class tiles
#define NIMG  8

// d_out layout (flat, return order): cls_score, distances, reps, emb
#define OFF_CLS   0
#define OFF_DIST  (NIMG*NCLS*HW)               // 10,616,832
#define OFF_REPS  (OFF_DIST + (size_t)NIMG*NCLS*HW)
#define OFF_EMB   (OFF_REPS + NCLS*EDIM)

// ---------------- Kernel 1: normalize class prototypes (+ zero-padded copy) ----------------
// grid 96: blocks 0..80 normalize; 81..95 zero-fill the padded copy in d_ws.
__global__ __launch_bounds__(256) void dml_reps_norm(
    const float* __restrict__ reps_raw,
    float* __restrict__ reps_out,          // [81,256] -> part of d_out
    float* __restrict__ reps_pad)          // [96,256] -> d_ws
{
  __shared__ float red[8];
  const int c = blockIdx.x;          // 0..95 (uniform per block)
  const int t = threadIdx.x;         // 0..255 == e
  if (c >= NCLS) {                   // pad rows: zeros so dot==0 for fake classes
    reps_pad[c * EDIM + t] = 0.f;
    return;
  }
  float v  = reps_raw[c * EDIM + t];
  float ss = v * v;
  #pragma unroll
  for (int o = 16; o >= 1; o >>= 1) ss += __shfl_xor(ss, o, 32);
  if ((t & 31) == 0) red[t >> 5] = ss;
  __syncthreads();
  if (t < 8) {
    float s = red[t];
    #pragma unroll
    for (int o = 4; o >= 1; o >>= 1) s += __shfl_xor(s, o, 8);
    if (t == 0) red[0] = s;
  }
  __syncthreads();
  const float inv = 1.0f / fmaxf(sqrtf(red[0]), 1e-12f);
  const float r = v * inv;
  reps_out[c * EDIM + t] = r;
  reps_pad[c * EDIM + t] = r;
}

// ---------------- Kernel 2: fused normalize + WMMA GEMM + epilogue ----------------
// grid: 512 blocks (8 images * 64 pixel-chunks), 256 threads (8 waves)
// block handles 256 pixels of one image; each wave does two 16-pixel tiles.
__global__ __launch_bounds__(256) void dml_main(
    const float* __restrict__ x,        // [N, E, HW]
    const float* __restrict__ reps_pad, // [96, 256] normalized, zero-padded
    float* __restrict__ cls,            // [N, 81, HW]
    float* __restrict__ dist,           // [N, 81, HW]
    float* __restrict__ emb)            // [N, E, HW]
{
  __shared__ float s_inv[256];
  __shared__ float s_stage[8][NPAD * 17];   // per-wave transpose stage (52,224 B)

  const int n       = blockIdx.x >> 6;
  const int pixbase = (blockIdx.x & 63) * 256;
  const int t       = threadIdx.x;
  const int wave    = t >> 5;
  const int lane    = t & 31;
  const int half    = lane >> 4;
  const int m16     = lane & 15;

  const float* xn = x + (size_t)n * EDIM * HW;

  // ---- Phase 1: per-pixel inverse L2 norm over E (coalesced) ----
  const int pix = pixbase + t;
  float ss = 0.f;
  #pragma unroll 8
  for (int e = 0; e < EDIM; ++e) {
    float v = xn[e * HW + pix];
    ss = fmaf(v, v, ss);
  }
  const float inv = 1.0f / fmaxf(sqrtf(ss), 1e-12f);
  s_inv[t] = inv;

  // ---- Phase 2: stream normalized embedding to d_out (coalesced) ----
  float* embn = emb + (size_t)n * EDIM * HW;
  #pragma unroll 8
  for (int e = 0; e < EDIM; ++e)
    embn[e * HW + pix] = xn[e * HW + pix] * inv;
  __syncthreads();

  // ---- Phase 3: WMMA tiles ----
  for (int tt = 0; tt < 2; ++tt) {
    const int tile  = wave * 2 + tt;
    const int tpix  = pixbase + tile * 16;     // tile's first pixel
    const int myPix = tpix + m16;              // A-row pixel for this lane
    const float rowInv = s_inv[tile * 16 + m16];
    const int kb_off = half * 2;               // K sub-offset per ISA f32 layout
    const float* brow = reps_pad + m16 * EDIM + kb_off;  // lane's B column base

    v8f acc[6] = {};
    #pragma unroll 2
    for (int k = 0; k < EDIM; k += 4) {
      const int kb = k + kb_off;
      v2f a;
      a.x = xn[(kb + 0) * HW + myPix] * rowInv;
      a.y = xn[(kb + 1) * HW + myPix] * rowInv;
      #pragma unroll
      for (int ct = 0; ct < 6; ++ct) {
        v2f b;
        b.x = brow[ct * 16 * EDIM + k + 0];    // unconditional: pad rows are zero
        b.y = brow[ct * 16 * EDIM + k + 1];
        acc[ct] = __builtin_amdgcn_wmma_f32_16x16x4_f32(
            false, a, false, b, (short)0, acc[ct], false, false);
      }
    }

    // ---- per-pixel sum of probs over valid classes ----
    float psum[8];
    #pragma unroll
    for (int j = 0; j < 8; ++j) psum[j] = 0.f;
    #pragma unroll
    for (int ct = 0; ct < 6; ++ct) {
      const bool valid = (ct * 16 + m16) < NCLS;   // mask pad classes (exp(-4)!=0)
      #pragma unroll
      for (int j = 0; j < 8; ++j) {
        float d  = acc[ct][j];
        float sq = fmaxf(2.f - 2.f * d, 0.f);
        float p  = __expf(-2.f * sq);              // sigma=0.5 -> exp(-sq/0.5)
        psum[j] += valid ? p : 0.f;
      }
    }
    #pragma unroll
    for (int j = 0; j < 8; ++j) {
      #pragma unroll
      for (int o = 8; o >= 1; o >>= 1)             // reduce across 16-lane half
        psum[j] += __shfl_xor(psum[j], o, 32);
    }
    float rtot[8];
    #pragma unroll
    for (int j = 0; j < 8; ++j) rtot[j] = 1.0f / psum[j];

    float* stg = &s_stage[wave][0];

    // ---- distances: stage (transpose) then coalesced store ----
    #pragma unroll
    for (int ct = 0; ct < 6; ++ct)
      #pragma unroll
      for (int j = 0; j < 8; ++j) {
        float d  = acc[ct][j];
        float sq = fmaxf(2.f - 2.f * d, 0.f);
        stg[(ct * 16 + m16) * 17 + (j + half * 8)] = sqrtf(sq);
      }
    __syncthreads();
    {
      const int cw = lane >> 4;
      #pragma unroll 4
      for (int it = 0; it < 48; ++it) {
        const int c = it * 2 + cw;
        if (c < NCLS)
          dist[((size_t)n * NCLS + c) * HW + tpix + m16] = stg[c * 17 + m16];
      }
    }
    __syncthreads();

    // ---- cls_score: stage then coalesced store ----
    #pragma unroll
    for (int ct = 0; ct < 6; ++ct)
      #pragma unroll
      for (int j = 0; j < 8; ++j) {
        float d  = acc[ct][j];
        float sq = fmaxf(2.f - 2.f * d, 0.f);
        float p  = __expf(-2.f * sq);
        stg[(ct * 16 + m16) * 17 + (j + half * 8)] = p * rtot[j];
      }
    __syncthreads();
    {
      const int cw = lane >> 4;
      #pragma unroll 4
      for (int it = 0; it < 48; ++it) {
        const int c = it * 2 + cw;
        if (c < NCLS)
          cls[((size_t)n * NCLS + c) * HW + tpix + m16] = stg[c * 17 + m16];
      }
    }
    __syncthreads();
  }
}

extern "C" void kernel_launch(void* const* d_in, const int* in_sizes, int n_in,
                              void* d_out, int out_size, void* d_ws, size_t ws_size,
                              hipStream_t stream) {
  const float* x        = (const float*)d_in[0];
  const float* reps_raw = (const float*)d_in[1];
  float* out  = (float*)d_out;
  float* cls  = out + OFF_CLS;
  float* dst  = out + OFF_DIST;
  float* reps = out + OFF_REPS;
  float* emb  = out + OFF_EMB;
  float* reps_pad = (float*)d_ws;              // 96*256*4 = 96 KB scratch

  dml_reps_norm<<<NPAD, 256, 0, stream>>>(reps_raw, reps, reps_pad);
  dml_main<<<NIMG * (HW / 256), 256, 0, stream>>>(x, reps_pad, cls, dst, emb);
}